// BilateralGrid_31894427140440
// MI455X (gfx1250) — compile-verified
//
#include <hip/hip_runtime.h>
#include <stdint.h>

// ---- problem constants (match reference) ----
#define H_IMG   1080
#define W_IMG   1920
#define HW      (H_IMG * W_IMG)
#define HG      16
#define WG_     16
#define LG      8
#define NC      12
#define ROWF    (WG_ * LG * NC)      // 1536 floats per grid y-row (6144 B)
#define GRID_ELEMS (HG * ROWF)       // 24576 floats per grid

// CDNA5 async global->LDS copy (ASYNCcnt path). GVS addressing:
//   mem = SADDR(64b sgpr pair) + VADDR(32b vgpr) ; LDS dst = 32b vgpr byte addr
__device__ __forceinline__ void async_ld_b128(uint32_t lds_byte, uint32_t goff,
                                              const float* sbase) {
  asm volatile("global_load_async_to_lds_b128 %0, %1, %2"
               :
               : "v"(lds_byte), "v"(goff), "s"(sbase)
               : "memory");
}

__device__ __forceinline__ void wait_async_zero() {
  asm volatile("s_wait_asynccnt 0" ::: "memory");
}

__global__ __launch_bounds__(256)
void bilateral_grid_apply(const float* __restrict__ rgb,
                          const float* __restrict__ grids,
                          const int*   __restrict__ idxp,
                          float*       __restrict__ out) {
  __shared__ alignas(16) float raw[2 * ROWF];   // rows y0, y1 of the grid
  __shared__ alignas(16) float gY[ROWF];        // y-interpolated row

  const int y   = blockIdx.x;
  const int tid = threadIdx.x;

  // --- block-uniform y interpolation setup (align_corners linspace) ---
  const float gy = (float)y * (15.0f / 1079.0f);
  const float fy = floorf(gy);
  const float wy = gy - fy;
  int y0 = (int)fy; y0 = y0 < 0 ? 0 : (y0 > 15 ? 15 : y0);
  int y1 = (y0 + 1 > 15) ? 15 : y0 + 1;

  const float* gbase = grids + (size_t)idxp[0] * GRID_ELEMS;
  const float* row0  = gbase + (size_t)y0 * ROWF;
  const float* row1  = gbase + (size_t)y1 * ROWF;

  // --- stage A: async DMA both grid rows into LDS (768 x b128 chunks) ---
  const uint32_t lds_base = (uint32_t)(uintptr_t)(&raw[0]);
  for (int i = tid; i < 768; i += 256) {
    if (i < 384) async_ld_b128(lds_base + (uint32_t)i * 16u,
                               (uint32_t)i * 16u, row0);
    else         async_ld_b128(lds_base + (uint32_t)i * 16u,
                               (uint32_t)(i - 384) * 16u, row1);
  }
  wait_async_zero();
  __syncthreads();

  // --- stage B: pre-lerp in y -> 6KB tile, collapses trilinear to bilinear ---
  const float wy0 = 1.0f - wy;
  for (int i = tid; i < ROWF; i += 256)
    gY[i] = wy0 * raw[i] + wy * raw[i + ROWF];
  __syncthreads();

  float* aff_out = out;                         // (H*W, 12)
  float* res_out = out + (size_t)HW * 12;       // (H*W, 3)

  // --- stage C: per-pixel slice + apply ---
  for (int x = tid; x < W_IMG; x += 256) {
    const int pix = y * W_IMG + x;
    const float r = rgb[pix];
    const float g = rgb[HW + pix];
    const float b = rgb[2 * HW + pix];

    float gray = r * 0.299f + g * 0.587f + b * 0.114f;
    gray = gray < 0.0f ? 0.0f : (gray > 1.0f ? 1.0f : gray);

    const float gz = gray * 7.0f;
    const float fz = floorf(gz);
    const float wz = gz - fz;
    int z0 = (int)fz; z0 = z0 < 0 ? 0 : (z0 > 7 ? 7 : z0);
    int z1 = (z0 + 1 > 7) ? 7 : z0 + 1;

    const float gx = (float)x * (15.0f / 1919.0f);
    const float fx = floorf(gx);
    const float wx = gx - fx;
    int x0 = (int)fx; x0 = x0 < 0 ? 0 : (x0 > 15 ? 15 : x0);
    int x1 = (x0 + 1 > 15) ? 15 : x0 + 1;

    const float w00 = (1.0f - wx) * (1.0f - wz);
    const float w01 = (1.0f - wx) * wz;
    const float w10 = wx * (1.0f - wz);
    const float w11 = wx * wz;

    // 12-float corners, 48B each -> 3 x ds_load_b128 per corner
    const float4* c00 = (const float4*)(gY + (x0 * LG + z0) * NC);
    const float4* c01 = (const float4*)(gY + (x0 * LG + z1) * NC);
    const float4* c10 = (const float4*)(gY + (x1 * LG + z0) * NC);
    const float4* c11 = (const float4*)(gY + (x1 * LG + z1) * NC);

    float a[12];
#pragma unroll
    for (int q = 0; q < 3; ++q) {
      const float4 v00 = c00[q], v01 = c01[q], v10 = c10[q], v11 = c11[q];
      a[q * 4 + 0] = w00 * v00.x + w01 * v01.x + w10 * v10.x + w11 * v11.x;
      a[q * 4 + 1] = w00 * v00.y + w01 * v01.y + w10 * v10.y + w11 * v11.y;
      a[q * 4 + 2] = w00 * v00.z + w01 * v01.z + w10 * v10.z + w11 * v11.z;
      a[q * 4 + 3] = w00 * v00.w + w01 * v01.w + w10 * v10.w + w11 * v11.w;
    }

    // affine coeffs: contiguous 48B -> 3 x global_store_b128, coalesced
    float4* ao = (float4*)(aff_out + (size_t)pix * 12);
    ao[0] = make_float4(a[0], a[1], a[2],  a[3]);
    ao[1] = make_float4(a[4], a[5], a[6],  a[7]);
    ao[2] = make_float4(a[8], a[9], a[10], a[11]);

    float* ro = res_out + (size_t)pix * 3;
    ro[0] = a[0] * r + a[1] * g + a[2]  * b + a[3];
    ro[1] = a[4] * r + a[5] * g + a[6]  * b + a[7];
    ro[2] = a[8] * r + a[9] * g + a[10] * b + a[11];
  }
}

extern "C" void kernel_launch(void* const* d_in, const int* in_sizes, int n_in,
                              void* d_out, int out_size, void* d_ws, size_t ws_size,
                              hipStream_t stream) {
  (void)in_sizes; (void)n_in; (void)out_size; (void)d_ws; (void)ws_size;
  const float* rgb   = (const float*)d_in[0];
  const float* grids = (const float*)d_in[1];
  const int*   idx   = (const int*)d_in[2];
  float*       out   = (float*)d_out;

  bilateral_grid_apply<<<H_IMG, 256, 0, stream>>>(rgb, grids, idx, out);
}